// SBG_83382495085286
// MI455X (gfx1250) — compile-verified
//
#include <hip/hip_runtime.h>
#include <hip/hip_bf16.h>

typedef __attribute__((ext_vector_type(2))) float v2f;
typedef __attribute__((ext_vector_type(8))) float v8f;

#define DA 128
#define DB 128

// ---------------------------------------------------------------------------
// GEMM: Y[m,n] = sum_k X[m,k] * W[k,n], X:[N,128], W:[128,128] row-major.
// One wave per 16x16 output tile, V_WMMA_F32_16X16X4_F32, K-loop of 32.
// A 16x4 layout: lanes 0-15 = rows, vgpr0/1 = K {0,1} (lanes<16) or {2,3}.
// B 4x16 layout: lanes 0-15 = cols, vgpr0/1 = K {0,1} (lanes<16) or {2,3}.
// C/D: vgpr j -> row m0 + j + 8*(lane>>4), col n0 + (lane&15).
// ---------------------------------------------------------------------------
__global__ __launch_bounds__(32) void gemm_xw(
    const float* __restrict__ X,
    const float* __restrict__ W0, const float* __restrict__ W1, const float* __restrict__ W2,
    float* __restrict__ Y0, float* __restrict__ Y1, float* __restrict__ Y2)
{
    const int lane = threadIdx.x;
    const int m0 = blockIdx.x * 16;
    const int n0 = blockIdx.y * 16;
    const float* __restrict__ W = (blockIdx.z == 0) ? W0 : (blockIdx.z == 1) ? W1 : W2;
    float* __restrict__ Y = (blockIdx.z == 0) ? Y0 : (blockIdx.z == 1) ? Y1 : Y2;

    const int mrow = m0 + (lane & 15);
    const int ncol = n0 + (lane & 15);
    const int khalf = (lane >> 4) * 2;   // 0 or 2

    v8f c = {};
#pragma unroll
    for (int k = 0; k < DA; k += 4) {
        v2f a, b;
        a[0] = X[mrow * DA + k + khalf + 0];
        a[1] = X[mrow * DA + k + khalf + 1];
        b[0] = W[(k + khalf + 0) * DB + ncol];
        b[1] = W[(k + khalf + 1) * DB + ncol];
        c = __builtin_amdgcn_wmma_f32_16x16x4_f32(false, a, false, b, (short)0, c,
                                                  false, false);
    }
#pragma unroll
    for (int j = 0; j < 8; ++j) {
        int r = m0 + j + (lane >> 4) * 8;
        Y[r * DB + ncol] = c[j];
    }
}

// ---------------------------------------------------------------------------
// SPMM scatter: Out[row[e], f] += w[e] * X[col[e], f]
// idx: [2,E] int32 (idx[0..E-1]=row, idx[E..2E-1]=col). 128 lanes per edge:
// coalesced 512B gather of X row, coalesced f32 atomics into Out row (L2).
// ---------------------------------------------------------------------------
__global__ void spmm_scatter(const int* __restrict__ idx, const float* __restrict__ w,
                             const float* __restrict__ X, float* __restrict__ Out, int E)
{
    int gid = blockIdx.x * blockDim.x + threadIdx.x;
    int e = gid >> 7;
    int f = gid & 127;
    if (e >= E) return;
    int r = idx[e];
    int col = idx[E + e];
    float v = X[col * DB + f] * w[e];
    atomicAdd(&Out[r * DB + f], v);
}

// dst = ca*A + cb*dst
__global__ void axpby(float* __restrict__ dst, const float* __restrict__ A,
                      float ca, float cb, int n)
{
    int i = blockIdx.x * blockDim.x + threadIdx.x;
    if (i < n) dst[i] = ca * A[i] + cb * dst[i];
}

// dst = c1*dst + c2*B + c3*C
__global__ void tricomb(float* __restrict__ dst, const float* __restrict__ B,
                        const float* __restrict__ C, float c1, float c2, float c3, int n)
{
    int i = blockIdx.x * blockDim.x + threadIdx.x;
    if (i < n) dst[i] = c1 * dst[i] + c2 * B[i] + c3 * C[i];
}

// ---------------------------------------------------------------------------
// Column sum / sum-of-squares for three [N,128] matrices -> sums[m*256 + f]
// (sum) and sums[m*256 + 128 + f] (sumsq). 256 threads, 256 rows per block.
// ---------------------------------------------------------------------------
__global__ void col_stats(const float* __restrict__ M0, const float* __restrict__ M1,
                          const float* __restrict__ M2, float* __restrict__ sums, int N)
{
    const float* __restrict__ M = (blockIdx.y == 0) ? M0 : (blockIdx.y == 1) ? M1 : M2;
    int f = threadIdx.x & 127;
    int sub = threadIdx.x >> 7;
    int r0 = blockIdx.x * 256 + sub;
    int rend = min(N, (int)blockIdx.x * 256 + 256);
    float s = 0.f, q = 0.f;
    for (int r = r0; r < rend; r += 2) {
        float v = M[r * DB + f];
        s += v; q += v * v;
    }
    __shared__ float ls[256], lq[256];
    ls[threadIdx.x] = s; lq[threadIdx.x] = q;
    __syncthreads();
    if (sub == 0) {
        s += ls[threadIdx.x + 128];
        q += lq[threadIdx.x + 128];
        atomicAdd(&sums[blockIdx.y * 256 + f], s);
        atomicAdd(&sums[blockIdx.y * 256 + 128 + f], q);
    }
}

// Fold mean/var/gamma/beta into per-column scale & shift.
__global__ void bn_finalize(const float* __restrict__ sums,
                            const float* __restrict__ g0, const float* __restrict__ b0,
                            const float* __restrict__ g1, const float* __restrict__ b1,
                            const float* __restrict__ g2, const float* __restrict__ b2,
                            float* __restrict__ scalesh, float invN)
{
    int i = threadIdx.x;
    if (i >= 384) return;
    int m = i >> 7, f = i & 127;
    float mean = sums[m * 256 + f] * invN;
    float var = sums[m * 256 + 128 + f] * invN - mean * mean;
    const float* g = (m == 0) ? g0 : (m == 1) ? g1 : g2;
    const float* be = (m == 0) ? b0 : (m == 1) ? b1 : b2;
    float sc = g[f] * rsqrtf(var + 1e-5f);
    scalesh[m * 256 + f] = sc;
    scalesh[m * 256 + 128 + f] = be[f] - mean * sc;
}

// ---------------------------------------------------------------------------
// Final GEMM: [N,384] (virtual concat, BN+PReLU fused into A-load) @ Wmlp[384,128].
// ---------------------------------------------------------------------------
__global__ __launch_bounds__(32) void gemm_final(
    const float* __restrict__ M0, const float* __restrict__ M1, const float* __restrict__ M2,
    const float* __restrict__ scalesh, const float* __restrict__ prelu_w,
    const float* __restrict__ Wmlp, float* __restrict__ Out)
{
    const int lane = threadIdx.x;
    const int m0 = blockIdx.x * 16;
    const int n0 = blockIdx.y * 16;
    const int mrow = m0 + (lane & 15);
    const int ncol = n0 + (lane & 15);
    const int khalf = (lane >> 4) * 2;
    const float pw = prelu_w[0];

    v8f c = {};
#pragma unroll 8
    for (int k = 0; k < 3 * DB; k += 4) {
        v2f a, b;
#pragma unroll
        for (int t = 0; t < 2; ++t) {
            int kk = k + khalf + t;
            int mm = kk >> 7, cc = kk & 127;
            const float* __restrict__ M = (mm == 0) ? M0 : (mm == 1) ? M1 : M2;
            float v = M[mrow * DB + cc] * scalesh[mm * 256 + cc]
                      + scalesh[mm * 256 + 128 + cc];
            v = (v >= 0.f) ? v : pw * v;
            a[t] = v;
            b[t] = Wmlp[kk * DB + ncol];
        }
        c = __builtin_amdgcn_wmma_f32_16x16x4_f32(false, a, false, b, (short)0, c,
                                                  false, false);
    }
#pragma unroll
    for (int j = 0; j < 8; ++j) {
        int r = m0 + j + (lane >> 4) * 8;
        Out[r * DB + ncol] = c[j];
    }
}

// Row L2 normalize in place: 2 rows per 256-thread block.
__global__ void l2norm(float* __restrict__ Out, int N)
{
    __shared__ float red[256];
    int f = threadIdx.x & 127;
    int sub = threadIdx.x >> 7;
    int r = blockIdx.x * 2 + sub;
    float v = (r < N) ? Out[r * DB + f] : 0.f;
    red[threadIdx.x] = v * v;
    __syncthreads();
#pragma unroll
    for (int s = 64; s > 0; s >>= 1) {
        if (f < s) red[threadIdx.x] += red[threadIdx.x + s];
        __syncthreads();
    }
    float nrm = fmaxf(sqrtf(red[sub * 128]), 1e-12f);
    if (r < N) Out[r * DB + f] = v / nrm;
}

extern "C" void kernel_launch(void* const* d_in, const int* in_sizes, int n_in,
                              void* d_out, int out_size, void* d_ws, size_t ws_size,
                              hipStream_t stream)
{
    const float* x          = (const float*)d_in[0];
    const int*   pos_index  = (const int*)d_in[1];
    const float* pos_weight = (const float*)d_in[2];
    const int*   neg_index  = (const int*)d_in[3];
    const float* neg_weight = (const float*)d_in[4];
    const float* W_org      = (const float*)d_in[5];
    const float* W_pos      = (const float*)d_in[6];
    const float* W_neg      = (const float*)d_in[7];
    const float* g_org      = (const float*)d_in[8];
    const float* b_org      = (const float*)d_in[9];
    const float* g_pos      = (const float*)d_in[10];
    const float* b_pos      = (const float*)d_in[11];
    const float* g_neg      = (const float*)d_in[12];
    const float* b_neg      = (const float*)d_in[13];
    const float* prelu_w    = (const float*)d_in[14];
    const float* W_mlp      = (const float*)d_in[15];
    float* out = (float*)d_out;

    const int N = in_sizes[0] / DA;       // 40000
    const int E = in_sizes[2];            // 640000
    const size_t SZ = (size_t)N * DB;     // elements per feature matrix

    float* ws      = (float*)d_ws;
    float* Yorg    = ws;                  // x @ W_org
    float* Ypos    = ws + SZ;             // x @ W_pos -> later: neg-final
    float* Yneg    = ws + 2 * SZ;         // x @ W_neg
    float* T       = ws + 3 * SZ;         // spmm scratch (x1)
    float* U       = ws + 4 * SZ;         // spmm scratch -> pos-final
    float* sums    = ws + 5 * SZ;         // 768 floats: per-col sum/sumsq x3
    float* scalesh = sums + 768;          // 768 floats: per-col scale/shift x3

    // Jacobi coefficients (k=3, a=b=1): x1 = a1*x0 + b1*spmm(x0);
    // x2 = c1*spmm(x1) + c2*x1 - c3*x0
    const double a = 1.0, b = 1.0; const int L = 2;
    const float a1 = (float)((a - b) / 2.0);
    const float b1 = (float)((a + b + 2.0) / 2.0);
    const float c1 = (float)((2 * L + a + b - 1) * (2 * L + a + b) / (2.0 * L) / (L + a + b));
    const float c2 = (float)((2 * L + a + b - 1) * (a * a - b * b) / (2.0 * L) / (L + a + b)
                             / (2 * L + a + b - 2));
    const float c3 = (float)((L + a) * (L + b) * (2 * L + a + b) / (double)L / (L + a + b)
                             / (2 * L + a + b - 2));

    // 1) three input GEMMs through WMMA f32
    dim3 gG(N / 16, DB / 16, 3);
    gemm_xw<<<gG, 32, 0, stream>>>(x, W_org, W_pos, W_neg, Yorg, Ypos, Yneg);

    const int eBlocks = (E * 128 + 255) / 256;
    const int nBlocks = (int)((SZ + 255) / 256);

    // 2) pos Jacobi chain: final lands in U
    hipMemsetAsync(T, 0, SZ * 4, stream);
    spmm_scatter<<<eBlocks, 256, 0, stream>>>(pos_index, pos_weight, Ypos, T, E);
    axpby<<<nBlocks, 256, 0, stream>>>(T, Ypos, a1, b1, (int)SZ);            // T = x1
    hipMemsetAsync(U, 0, SZ * 4, stream);
    spmm_scatter<<<eBlocks, 256, 0, stream>>>(pos_index, pos_weight, T, U, E);
    tricomb<<<nBlocks, 256, 0, stream>>>(U, T, Ypos, c1, c2, -c3, (int)SZ);  // U = x2

    // 3) neg Jacobi chain: final lands in Ypos buffer (recycled)
    hipMemsetAsync(T, 0, SZ * 4, stream);
    spmm_scatter<<<eBlocks, 256, 0, stream>>>(neg_index, neg_weight, Yneg, T, E);
    axpby<<<nBlocks, 256, 0, stream>>>(T, Yneg, a1, b1, (int)SZ);
    hipMemsetAsync(Ypos, 0, SZ * 4, stream);
    spmm_scatter<<<eBlocks, 256, 0, stream>>>(neg_index, neg_weight, T, Ypos, E);
    tricomb<<<nBlocks, 256, 0, stream>>>(Ypos, T, Yneg, c1, c2, -c3, (int)SZ);

    // 4) batchnorm statistics + fold
    hipMemsetAsync(sums, 0, 768 * 4, stream);
    dim3 gS((N + 255) / 256, 3);
    col_stats<<<gS, 256, 0, stream>>>(Yorg, U, Ypos, sums, N);
    bn_finalize<<<1, 384, 0, stream>>>(sums, g_org, b_org, g_pos, b_pos, g_neg, b_neg,
                                       scalesh, 1.0f / (float)N);

    // 5) fused BN+PReLU+concat @ W_mlp via WMMA, then row L2 normalize
    dim3 gF(N / 16, DB / 16);
    gemm_final<<<gF, 32, 0, stream>>>(Yorg, U, Ypos, scalesh, prelu_w, W_mlp, out);
    l2norm<<<(N + 1) / 2, 256, 0, stream>>>(out, N);
}